// RGCN_72258529788422
// MI455X (gfx1250) — compile-verified
//
#include <hip/hip_runtime.h>

typedef __attribute__((ext_vector_type(2))) float v2f;
typedef __attribute__((ext_vector_type(8))) float v8f;

#define HF     128   // hidden dim
#define NGRAPH 256   // number of graphs

// ---------------------------------------------------------------- utilities
__global__ void zero_f32(float* __restrict__ p, int n) {
    int i = blockIdx.x * blockDim.x + threadIdx.x;
    if (i < n) p[i] = 0.0f;
}

__global__ void deg_init(float* __restrict__ deg, int n) {
    int i = blockIdx.x * blockDim.x + threadIdx.x;
    if (i < n) deg[i] = 1.0f;               // self loop contributes 1
}

__global__ void deg_count(const int* __restrict__ dst, float* __restrict__ deg, int e) {
    int i = blockIdx.x * blockDim.x + threadIdx.x;
    if (i < e) atomicAdd(&deg[dst[i]], 1.0f);
}

__global__ void deg_rsqrt(float* __restrict__ d, int n) {
    int i = blockIdx.x * blockDim.x + threadIdx.x;
    if (i < n) d[i] = rsqrtf(d[i]);         // deg >= 1 always
}

// ------------------------------------------------------- fp32 WMMA GEMM
// H[m, 0:128] = X[m, 0:K] @ W[0:K, 0:128]
// block = 256 threads (8 waves); block -> 16-row M tile; wave w -> N tile [16w,16w+16)
__global__ void gemm_wmma(const float* __restrict__ X, const float* __restrict__ W,
                          float* __restrict__ Hout, int n, int K) {
    __shared__ float sA[16 * 32];        // 16 rows x 32 k
    __shared__ float sB[32 * HF];        // 32 k x 128 n

    const int tid  = threadIdx.x;
    const int wave = tid >> 5;
    const int lane = tid & 31;
    const int m0   = blockIdx.x * 16;
    const int n0   = wave * 16;
    const int ml   = lane & 15;           // A: row index / B: col index
    const int koff = (lane >> 4) << 1;    // upper half-wave handles K+2,K+3

    v8f acc = {0.f, 0.f, 0.f, 0.f, 0.f, 0.f, 0.f, 0.f};

    for (int k0 = 0; k0 < K; k0 += 32) {
        // stage A chunk (zero-pad K tail, clamp M for safety)
        for (int i = tid; i < 16 * 32; i += 256) {
            int r = i >> 5, c = i & 31;
            int mr = m0 + r; if (mr >= n) mr = n - 1;
            int kc = k0 + c;
            sA[i] = (kc < K) ? X[(size_t)mr * K + kc] : 0.0f;
        }
        // stage B chunk (zero-pad K tail)
        for (int i = tid; i < 32 * HF; i += 256) {
            int r = i >> 7, c = i & (HF - 1);
            int kr = k0 + r;
            sB[i] = (kr < K) ? W[(size_t)kr * HF + c] : 0.0f;
        }
        __syncthreads();

#pragma unroll
        for (int kk = 0; kk < 32; kk += 4) {
            v2f a, b;
            a[0] = sA[ml * 32 + kk + koff];
            a[1] = sA[ml * 32 + kk + koff + 1];
            b[0] = sB[(kk + koff) * HF + n0 + ml];
            b[1] = sB[(kk + koff + 1) * HF + n0 + ml];
            acc = __builtin_amdgcn_wmma_f32_16x16x4_f32(
                false, a, false, b, (short)0, acc, false, false);
        }
        __syncthreads();
    }

    // C/D layout: lane holds column n0+ml; VGPR v holds row v + 8*(lane>=16)
    const int mhi = (lane >> 4) * 8;
    const int nn  = n0 + ml;
#pragma unroll
    for (int v = 0; v < 8; v++) {
        int m = m0 + mhi + v;
        if (m < n) Hout[(size_t)m * HF + nn] = acc[v];
    }
}

// ------------------------------------------------- aggregation (scatter)
// agg[i] = h[i] * dinv[i]^2   (self loop; also deterministic init of agg)
__global__ void self_init(const float* __restrict__ h, const float* __restrict__ dinv,
                          float* __restrict__ agg, int n) {
    int gid  = blockIdx.x * blockDim.x + threadIdx.x;
    int node = gid >> 5;
    int lane = gid & 31;
    if (node >= n) return;
    float w = dinv[node] * dinv[node];
    float4 hv = ((const float4*)(h + (size_t)node * HF))[lane];
    float4 o;
    o.x = hv.x * w; o.y = hv.y * w; o.z = hv.z * w; o.w = hv.w * w;
    ((float4*)(agg + (size_t)node * HF))[lane] = o;
}

// one wave per edge: agg[dst] += h[src] * dinv[src]*dinv[dst]
__global__ void edge_scatter(const int* __restrict__ src, const int* __restrict__ dst,
                             const float* __restrict__ dinv, const float* __restrict__ h,
                             float* __restrict__ agg, int e) {
    int gid  = blockIdx.x * blockDim.x + threadIdx.x;
    int edge = gid >> 5;
    int lane = gid & 31;
    if (edge >= e) return;
    int s = src[edge], d = dst[edge];
    float nrm = dinv[s] * dinv[d];
    float4 hv = ((const float4*)(h + (size_t)s * HF))[lane];
    float* ap = agg + (size_t)d * HF + lane * 4;
    atomicAdd(ap + 0, hv.x * nrm);
    atomicAdd(ap + 1, hv.y * nrm);
    atomicAdd(ap + 2, hv.z * nrm);
    atomicAdd(ap + 3, hv.w * nrm);
}

// -------------------------------------------------------- pooling + head
// one wave per node: pool[batch[node]] += relu(agg[node] + bias); cnt[g] += 1
__global__ void pool_accum(const float* __restrict__ agg, const float* __restrict__ bias,
                           const int* __restrict__ batch, int n,
                           float* __restrict__ pool, float* __restrict__ cnt) {
    int gid  = blockIdx.x * blockDim.x + threadIdx.x;
    int node = gid >> 5;
    int lane = gid & 31;
    if (node >= n) return;
    int g = batch[node];
    float4 hv = ((const float4*)(agg + (size_t)node * HF))[lane];
    float4 bv = ((const float4*)bias)[lane];
    float* pp = pool + (size_t)g * HF + lane * 4;
    atomicAdd(pp + 0, fmaxf(hv.x + bv.x, 0.0f));
    atomicAdd(pp + 1, fmaxf(hv.y + bv.y, 0.0f));
    atomicAdd(pp + 2, fmaxf(hv.z + bv.z, 0.0f));
    atomicAdd(pp + 3, fmaxf(hv.w + bv.w, 0.0f));
    if (lane == 0) atomicAdd(&cnt[g], 1.0f);
}

// out[g, c] = mean(pool[g]) @ W_lin[:, c] + b_lin[c]; one wave per graph
__global__ void final_lin(const float* __restrict__ pool, const float* __restrict__ cnt,
                          const float* __restrict__ Wl, const float* __restrict__ bl,
                          float* __restrict__ out) {
    int g    = blockIdx.x;
    int lane = threadIdx.x;   // 32 lanes
    float c  = fmaxf(cnt[g], 1.0f);
    float s0 = 0.0f, s1 = 0.0f;
    for (int j = lane; j < HF; j += 32) {
        float m = pool[(size_t)g * HF + j] / c;
        s0 += m * Wl[j * 2 + 0];
        s1 += m * Wl[j * 2 + 1];
    }
    for (int off = 16; off > 0; off >>= 1) {
        s0 += __shfl_down(s0, off, 32);
        s1 += __shfl_down(s1, off, 32);
    }
    if (lane == 0) {
        out[g * 2 + 0] = s0 + bl[0];
        out[g * 2 + 1] = s1 + bl[1];
    }
}

// ------------------------------------------------------------- launcher
extern "C" void kernel_launch(void* const* d_in, const int* in_sizes, int n_in,
                              void* d_out, int out_size, void* d_ws, size_t ws_size,
                              hipStream_t stream) {
    const float* p_x   = (const float*)d_in[0];
    const int*   p_ei  = (const int*)  d_in[1];
    const float* r_x   = (const float*)d_in[2];
    const int*   r_ei  = (const int*)  d_in[3];
    const int*   batch = (const int*)  d_in[4];
    const float* W_p   = (const float*)d_in[5];
    const float* b_p   = (const float*)d_in[6];
    const float* W_r   = (const float*)d_in[7];
    const float* b_r   = (const float*)d_in[8];
    const float* W_lin = (const float*)d_in[9];
    const float* b_lin = (const float*)d_in[10];

    const int Kp = 1280, Kr = 120;
    const int Np = in_sizes[0] / Kp;
    const int Ep = in_sizes[1] / 2;
    const int Nr = in_sizes[2] / Kr;
    const int Er = in_sizes[3] / 2;

    const size_t maxN = (size_t)(Np > Nr ? Np : Nr);
    float* dinv = (float*)d_ws;                 // maxN
    float* hbuf = dinv + maxN;                  // maxN * HF
    float* agg  = hbuf + maxN * HF;             // maxN * HF
    float* pool = agg  + maxN * HF;             // NGRAPH * HF
    float* cnt  = pool + (size_t)NGRAPH * HF;   // NGRAPH (contiguous after pool)

    const int B = 256;

    // zero pool + cnt (contiguous)
    zero_f32<<<(NGRAPH * HF + NGRAPH + B - 1) / B, B, 0, stream>>>(
        pool, NGRAPH * HF + NGRAPH);

    auto phase = [&](const float* X, int K, const float* W, const float* bias,
                     const int* ei, int E, int n, const int* batch_part) {
        deg_init <<<(n + B - 1) / B, B, 0, stream>>>(dinv, n);
        deg_count<<<(E + B - 1) / B, B, 0, stream>>>(ei + E, dinv, E);
        deg_rsqrt<<<(n + B - 1) / B, B, 0, stream>>>(dinv, n);
        gemm_wmma<<<(n + 15) / 16, 256, 0, stream>>>(X, W, hbuf, n, K);
        self_init<<<(unsigned)(((size_t)n * 32 + B - 1) / B), B, 0, stream>>>(
            hbuf, dinv, agg, n);
        edge_scatter<<<(unsigned)(((size_t)E * 32 + B - 1) / B), B, 0, stream>>>(
            ei, ei + E, dinv, hbuf, agg, E);
        pool_accum<<<(unsigned)(((size_t)n * 32 + B - 1) / B), B, 0, stream>>>(
            agg, bias, batch_part, n, pool, cnt);
    };

    phase(p_x, Kp, W_p, b_p, p_ei, Ep, Np, batch);
    phase(r_x, Kr, W_r, b_r, r_ei, Er, Nr, batch + Np);

    final_lin<<<NGRAPH, 32, 0, stream>>>(pool, cnt, W_lin, b_lin, (float*)d_out);
}